// Weighted3DTransformNormLoss_39994735460545
// MI455X (gfx1250) — compile-verified
//
#include <hip/hip_runtime.h>
#include <hip/hip_bf16.h>

// ---------------------------------------------------------------------------
// Weighted 3D transform + norm loss, MI455X (gfx1250, wave32).
// Memory-bound streaming reduction: 56 B/pixel, 137.6 MB total -> ~5.9us at
// 23.3 TB/s HBM. Strategy: float4 (b128) coalesced loads, per-thread fp32
// accumulation, wave reduction via V_WMMA_F32_16X16X4_F32 (exact fp32),
// deterministic two-pass final reduction through d_ws.
// ---------------------------------------------------------------------------

typedef __attribute__((ext_vector_type(2))) float v2f;
typedef __attribute__((ext_vector_type(8))) float v8f;

constexpr int Bc = 32;
constexpr int Kc = 8;
constexpr int Hc = 240;
constexpr int Wc = 320;
constexpr int HW = Hc * Wc;                 // 76800
constexpr int PIX_PER_THREAD = 4;
constexpr int BLOCK = 256;
constexpr int GROUPS_PER_B = HW / PIX_PER_THREAD;        // 19200
constexpr int BLOCKS_PER_B = GROUPS_PER_B / BLOCK;       // 75
constexpr int NUM_PARTIALS = BLOCKS_PER_B * Bc;          // 2400
constexpr float EPSc = 0.001f;
constexpr float SCALE = 0.5f / (float)(Bc * 3 * Hc * Wc); // 0.5 / inputpts.size

static_assert(HW % (PIX_PER_THREAD * BLOCK) == 0, "exact tiling required");

// Exact fp32 reduction of 32 lane values using one WMMA.
// A layout (16x4 f32, documented): VGPR0 = {K=0 for lanes 0-15 (M=lane),
// K=2 for lanes 16-31 (M=lane-16)}, VGPR1 = {K=1, K=3}. With A[.][0]=x,
// A[.][1..3 via VGPR1]=0 and B = ones, D[m][n] = x_m + x_{m+16} regardless of
// B's lane layout. D VGPR r: lanes 0-15 hold row r, lanes 16-31 hold row r+8.
__device__ __forceinline__ float wave_reduce_wmma(float x) {
  v2f a; a[0] = x;    a[1] = 0.0f;
  v2f b; b[0] = 1.0f; b[1] = 1.0f;
  v8f c = {};
  v8f d = __builtin_amdgcn_wmma_f32_16x16x4_f32(
      /*neg_a=*/false, a, /*neg_b=*/false, b,
      /*c_mod=*/(short)0, c, /*reuse_a=*/false, /*reuse_b=*/false);
  float local = ((d[0] + d[1]) + (d[2] + d[3])) + ((d[4] + d[5]) + (d[6] + d[7]));
  // lanes 0-15 hold sum of rows 0-7 (= x0..7 + x16..23), lanes 16-31 the rest
  return local + __shfl_xor(local, 16, 32);
}

__global__ void __launch_bounds__(BLOCK)
wloss_partial(const float* __restrict__ pts,
              const float* __restrict__ masks,
              const float* __restrict__ tfms,
              const float* __restrict__ flows,
              float* __restrict__ partials) {
  __shared__ float sT[Kc * 12];   // per-k: [a00 a01 a02 t0; a10..t1; a20..t2], A = R - I
  __shared__ float sWave[BLOCK / 32];

  const int b   = blockIdx.y;
  const int tid = threadIdx.x;

  if (tid < Kc * 12) {
    float v = tfms[b * Kc * 12 + tid];
    const int r = (tid % 12) / 4;
    const int col = tid % 4;
    if (col == r) v -= 1.0f;      // subtract identity on the 3x3 diagonal
    sT[tid] = v;
  }
  __syncthreads();

  const int n = (blockIdx.x * BLOCK + tid) * PIX_PER_THREAD;  // < HW by construction
  const float* pb = pts   + (size_t)b * 3  * HW;
  const float* fb = flows + (size_t)b * 3  * HW;
  const float* mb = masks + (size_t)b * Kc * HW;

  const float4 px = *(const float4*)(pb + 0 * HW + n);
  const float4 py = *(const float4*)(pb + 1 * HW + n);
  const float4 pz = *(const float4*)(pb + 2 * HW + n);
  const float4 fx = *(const float4*)(fb + 0 * HW + n);
  const float4 fy = *(const float4*)(fb + 1 * HW + n);
  const float4 fz = *(const float4*)(fb + 2 * HW + n);

  float4 mk[Kc];
#pragma unroll
  for (int k = 0; k < Kc; ++k)
    mk[k] = *(const float4*)(mb + k * HW + n);

  const float X[4] = {px.x, px.y, px.z, px.w};
  const float Y[4] = {py.x, py.y, py.z, py.w};
  const float Z[4] = {pz.x, pz.y, pz.z, pz.w};
  const float FX[4] = {fx.x, fx.y, fx.z, fx.w};
  const float FY[4] = {fy.x, fy.y, fy.z, fy.w};
  const float FZ[4] = {fz.x, fz.y, fz.z, fz.w};

  float partial = 0.0f;
#pragma unroll
  for (int i = 0; i < 4; ++i) {
    float msum = 0.0f;
#pragma unroll
    for (int k = 0; k < Kc; ++k) {
      const float* A = &sT[k * 12];
      const float d0 = fmaf(A[0], X[i], fmaf(A[1], Y[i], fmaf(A[2],  Z[i], A[3]  - FX[i])));
      const float d1 = fmaf(A[4], X[i], fmaf(A[5], Y[i], fmaf(A[6],  Z[i], A[7]  - FY[i])));
      const float d2 = fmaf(A[8], X[i], fmaf(A[9], Y[i], fmaf(A[10], Z[i], A[11] - FZ[i])));
      const float sq = fmaf(d0, d0, fmaf(d1, d1, d2 * d2));
      const float m  = ((const float*)&mk[k])[i];
      msum = fmaf(m, sq, msum);
    }
    const float sigma = fmaf(0.5f,
                             fmaf(FX[i], FX[i], fmaf(FY[i], FY[i], FZ[i] * FZ[i])),
                             EPSc);
    partial += msum / sigma;
  }

  // Wave reduction (WMMA) + deterministic block reduction.
  const float wsum = wave_reduce_wmma(partial);
  const int wid  = tid >> 5;
  const int lane = tid & 31;
  if (lane == 0) sWave[wid] = wsum;
  __syncthreads();
  if (tid == 0) {
    float s = 0.0f;
#pragma unroll
    for (int i = 0; i < BLOCK / 32; ++i) s += sWave[i];
    partials[blockIdx.y * gridDim.x + blockIdx.x] = s;
  }
}

__global__ void __launch_bounds__(BLOCK)
wloss_final(const float* __restrict__ partials, float* __restrict__ out) {
  __shared__ float sWave[BLOCK / 32];
  const int tid = threadIdx.x;

  float acc = 0.0f;
  for (int i = tid; i < NUM_PARTIALS; i += BLOCK)   // fixed order -> deterministic
    acc += partials[i];

  const float wsum = wave_reduce_wmma(acc);
  const int wid  = tid >> 5;
  const int lane = tid & 31;
  if (lane == 0) sWave[wid] = wsum;
  __syncthreads();
  if (tid == 0) {
    float s = 0.0f;
#pragma unroll
    for (int i = 0; i < BLOCK / 32; ++i) s += sWave[i];
    out[0] = s * SCALE;
  }
}

extern "C" void kernel_launch(void* const* d_in, const int* in_sizes, int n_in,
                              void* d_out, int out_size, void* d_ws, size_t ws_size,
                              hipStream_t stream) {
  const float* pts   = (const float*)d_in[0];  // (32, 3, 240, 320)
  const float* masks = (const float*)d_in[1];  // (32, 8, 240, 320)
  const float* tfms  = (const float*)d_in[2];  // (32, 8, 3, 4)
  const float* flows = (const float*)d_in[3];  // (32, 3, 240, 320)
  float* out      = (float*)d_out;
  float* partials = (float*)d_ws;              // NUM_PARTIALS floats

  dim3 grid(BLOCKS_PER_B, Bc);                 // 75 x 32
  wloss_partial<<<grid, BLOCK, 0, stream>>>(pts, masks, tfms, flows, partials);
  wloss_final<<<1, BLOCK, 0, stream>>>(partials, out);
}